// LiquidNeuronCUDA_53755810676810
// MI455X (gfx1250) — compile-verified
//
#include <hip/hip_runtime.h>
#include <math.h>

// ---------------------------------------------------------------------------
// Liquid (CT-RNN) neuron scan for MI455X (gfx1250), f32 WMMA path.
//   BATCH=64, SEQ=512, IN=512, HID=1024, DT=0.1
// Phase 0 (prep):   pack W_rec / W_in into K-major float2 pairs, copy h_init,
//                   zero grid-barrier counters.
// Phase 1 (proj):   xw[t][b][h] = x[b][t][:] . W_in[h][:] + bias[h]
//                   via V_WMMA_F32_16X16X4_F32, async-LDS-staged A tiles.
// Phase 2 (scan):   persistent kernel, 32 blocks x 256 threads (256 waves),
//                   one 16x16 h-tile per wave, grid barrier per timestep,
//                   double-buffered h in global (L2-resident), h tile staged
//                   with GLOBAL_LOAD_ASYNC_TO_LDS_B64 (ASYNCcnt path).
// ---------------------------------------------------------------------------

typedef __attribute__((ext_vector_type(2))) float v2f;
typedef __attribute__((ext_vector_type(8))) float v8f;

#define BATCH 64
#define SEQ   512
#define INDIM 512
#define HID   1024
#define DTC   0.1f

// ----------------------------- workspace layout ----------------------------
#define XW_BYTES    ((size_t)SEQ * BATCH * HID * 4)   // 134,217,728
#define HBUF_BYTES  ((size_t)BATCH * HID * 4)         // 262,144
#define WRECP_BYTES ((size_t)HID * HID * 4)           // 4,194,304
#define WINP_BYTES  ((size_t)HID * INDIM * 4)         // 2,097,152

#define OFF_XW    ((size_t)0)
#define OFF_HB0   (OFF_XW + XW_BYTES)
#define OFF_HB1   (OFF_HB0 + HBUF_BYTES)
#define OFF_WRECP (OFF_HB1 + HBUF_BYTES)
#define OFF_WINP  (OFF_WRECP + WRECP_BYTES)
#define OFF_BAR   (OFF_WINP + WINP_BYTES)

// ---------------------- CDNA5 async global->LDS copy -----------------------
// global_load_async_to_lds_b64: DMA 8 bytes memory -> LDS, no VGPR data,
// tracked with ASYNCcnt. vdst VGPR holds per-lane LDS byte address (low 32
// bits of a generic shared pointer == LDS offset per the aperture rules).
__device__ __forceinline__ void async_g2l_b64(unsigned lds_byte_addr,
                                              const void* gaddr) {
  asm volatile("global_load_async_to_lds_b64 %0, %1, off"
               :
               : "v"(lds_byte_addr), "v"((unsigned long long)(uintptr_t)gaddr)
               : "memory");
}

__device__ __forceinline__ void wait_async_zero() {
  asm volatile("s_wait_asynccnt 0x0" ::: "memory");
}

// --------------------------- grid barrier (device) -------------------------
__device__ __forceinline__ void grid_barrier(unsigned* cnt, unsigned* gen,
                                             unsigned nblk) {
  __syncthreads();
  if (threadIdx.x == 0) {
    __threadfence();  // make this block's h stores visible device-wide
    unsigned g = __hip_atomic_load(gen, __ATOMIC_RELAXED, __HIP_MEMORY_SCOPE_AGENT);
    unsigned arrived =
        __hip_atomic_fetch_add(cnt, 1u, __ATOMIC_ACQ_REL, __HIP_MEMORY_SCOPE_AGENT);
    if (arrived == nblk - 1u) {
      __hip_atomic_store(cnt, 0u, __ATOMIC_RELAXED, __HIP_MEMORY_SCOPE_AGENT);
      __hip_atomic_fetch_add(gen, 1u, __ATOMIC_RELEASE, __HIP_MEMORY_SCOPE_AGENT);
    } else {
      while (__hip_atomic_load(gen, __ATOMIC_ACQUIRE, __HIP_MEMORY_SCOPE_AGENT) == g) {
        __builtin_amdgcn_s_sleep(1);
      }
    }
  }
  __syncthreads();
}

// ------------------------------- prep kernel -------------------------------
__global__ void liquid_prep(const float* __restrict__ h_init,
                            const float* __restrict__ w_rec,
                            const float* __restrict__ w_in,
                            float* __restrict__ hb0,
                            v2f* __restrict__ wrecP,
                            v2f* __restrict__ winP,
                            unsigned* __restrict__ bar) {
  int i = blockIdx.x * blockDim.x + threadIdx.x;  // 0 .. 524287
  if (i < 2) bar[i] = 0u;
  if (i < BATCH * HID) hb0[i] = h_init[i];
  {
    int k2 = i >> 10, n = i & (HID - 1);
    if (k2 < (HID / 2)) {  // W_rec: 512 * 1024 pairs
      v2f v;
      v.x = w_rec[n * HID + 2 * k2];
      v.y = w_rec[n * HID + 2 * k2 + 1];
      wrecP[(size_t)k2 * HID + n] = v;
    }
    if (k2 < (INDIM / 2)) {  // W_in: 256 * 1024 pairs
      v2f v;
      v.x = w_in[n * INDIM + 2 * k2];
      v.y = w_in[n * INDIM + 2 * k2 + 1];
      winP[(size_t)k2 * HID + n] = v;
    }
  }
}

// --------------------------- phase 1: input proj ---------------------------
// Grid: 16384 blocks x 256 threads.  Block = one 16-row M tile (fixed t,
// 16 consecutive batches) x 128 output cols (8 waves x 16).
__global__ void liquid_input_proj(const float* __restrict__ x,
                                  const v2f* __restrict__ winP,
                                  const float* __restrict__ bias,
                                  float* __restrict__ xw) {
  __shared__ v2f ldsA[(INDIM / 2) * 16];  // 32 KB, [k2][row] swizzled

  const int bid  = blockIdx.x;
  const int Mt   = bid >> 3;            // 0..2047
  const int t    = Mt >> 2;             // timestep
  const int b0   = (Mt & 3) << 4;       // batch row offset
  const int wave = threadIdx.x >> 5;
  const int lane = threadIdx.x & 31;
  const int n0   = (((bid & 7) << 3) + wave) << 4;  // output col offset
  const int m    = lane & 15;           // A row / B,C column within tile
  const int mh   = lane >> 4;           // K-half selector
  const unsigned ldsAb = (unsigned)(uintptr_t)&ldsA[0];

  // Stage A tile: x[b0..b0+16)[t][0..512) -> LDS via async DMA, pair-packed
  // K-major, swizzled for conflict-free ds_load_b64 A-fragment reads.
  for (int p = threadIdx.x; p < 16 * (INDIM / 2); p += 256) {
    int row = p >> 8, k2 = p & ((INDIM / 2) - 1);
    const float* s = x + ((size_t)(b0 + row) * SEQ + t) * INDIM + 2 * k2;
    unsigned dst = ldsAb + (unsigned)((k2 * 16 + ((row + k2) & 15)) * 8);
    async_g2l_b64(dst, s);
  }
  wait_async_zero();
  __syncthreads();

  v8f c = {0.f, 0.f, 0.f, 0.f, 0.f, 0.f, 0.f, 0.f};
#pragma unroll 4
  for (int k2 = 0; k2 < INDIM / 2; k2 += 2) {  // K advances 4 per WMMA
    int kp = k2 + mh;
    v2f a = ldsA[kp * 16 + ((m + kp) & 15)];
    v2f b = winP[(size_t)kp * HID + n0 + m];
    c = __builtin_amdgcn_wmma_f32_16x16x4_f32(false, a, false, b,
                                              (short)0, c, false, false);
  }

  const float bs = bias[n0 + m];
  const size_t base = ((size_t)t * BATCH + b0) * HID + n0 + m;
#pragma unroll
  for (int r = 0; r < 8; ++r)
    xw[base + (size_t)(mh * 8 + r) * HID] = c[r] + bs;
}

// ---------------------------- phase 2: the scan ----------------------------
// Grid: 32 blocks x 256 threads (persistent). Block = 16 batch rows x 128
// hidden cols; wave = one 16x16 tile.  h double-buffered in global (L2).
__global__ void liquid_scan(const float* __restrict__ xw,
                            const v2f* __restrict__ wrecP,
                            const float* __restrict__ tau,
                            float* hb0, float* hb1,
                            float* __restrict__ out,
                            unsigned* __restrict__ bar) {
  __shared__ v2f ldsH[(HID / 2) * 16];  // 64 KB, [k2][row] swizzled

  const int bid  = blockIdx.x;
  const int b0   = (bid >> 3) << 4;     // batch row offset (0,16,32,48)
  const int wave = threadIdx.x >> 5;
  const int lane = threadIdx.x & 31;
  const int n0   = (((bid & 7) << 3) + wave) << 4;  // hidden col offset
  const int m    = lane & 15;
  const int mh   = lane >> 4;
  const float taun = tau[n0 + m];
  const unsigned ldsHb = (unsigned)(uintptr_t)&ldsH[0];

  for (int t = 0; t < SEQ; ++t) {
    const float* hc = (t & 1) ? hb1 : hb0;
    float* hn = (t == SEQ - 1) ? out : ((t & 1) ? hb0 : hb1);

    // Stage h tile (16 rows x 1024) -> LDS via async DMA (ASYNCcnt),
    // pair-packed K-major, swizzled.
    for (int p = threadIdx.x; p < 16 * (HID / 2); p += 256) {
      int row = p >> 9, k2 = p & ((HID / 2) - 1);
      const float* s = hc + (size_t)(b0 + row) * HID + 2 * k2;
      unsigned dst = ldsHb + (unsigned)((k2 * 16 + ((row + k2) & 15)) * 8);
      async_g2l_b64(dst, s);
    }
    wait_async_zero();
    __syncthreads();

    // C initialized with the precomputed input projection (bias folded in).
    v8f c;
    const size_t xbase = ((size_t)t * BATCH + b0) * HID + n0 + m;
#pragma unroll
    for (int r = 0; r < 8; ++r) c[r] = xw[xbase + (size_t)(mh * 8 + r) * HID];

#pragma unroll 4
    for (int k2 = 0; k2 < HID / 2; k2 += 2) {  // K advances 4 per WMMA
      int kp = k2 + mh;
      int kpre = (kp + 16 < HID / 2) ? kp + 16 : kp;
      __builtin_prefetch(&wrecP[(size_t)kpre * HID + n0 + m], 0, 0);
      v2f a = ldsH[kp * 16 + ((m + kp) & 15)];
      v2f b = wrecP[(size_t)kp * HID + n0 + m];
      c = __builtin_amdgcn_wmma_f32_16x16x4_f32(false, a, false, b,
                                                (short)0, c, false, false);
    }

    // Liquid Euler update: h' = h + DT*(tanh(pre) - h)/tau
#pragma unroll
    for (int r = 0; r < 8; ++r) {
      int row = mh * 8 + r;
      int col = n0 + m;
      int k2c = col >> 1;
      v2f hv = ldsH[k2c * 16 + ((row + k2c) & 15)];
      float hold = (col & 1) ? hv.y : hv.x;
      float hnew = hold + DTC * (tanhf(c[r]) - hold) / taun;
      hn[(size_t)(b0 + row) * HID + col] = hnew;
    }

    grid_barrier(bar, bar + 1, gridDim.x);
  }
}

// ------------------------------- launcher ----------------------------------
extern "C" void kernel_launch(void* const* d_in, const int* in_sizes, int n_in,
                              void* d_out, int out_size, void* d_ws,
                              size_t ws_size, hipStream_t stream) {
  (void)in_sizes; (void)n_in; (void)out_size; (void)ws_size;

  const float* x     = (const float*)d_in[0];  // [64][512][512]
  const float* h0    = (const float*)d_in[1];  // [64][1024]
  const float* w_in  = (const float*)d_in[2];  // [1024][512]
  const float* w_rec = (const float*)d_in[3];  // [1024][1024]
  const float* bias  = (const float*)d_in[4];  // [1024]
  const float* tau   = (const float*)d_in[5];  // [1024]

  char* ws = (char*)d_ws;
  float*    xw    = (float*)(ws + OFF_XW);
  float*    hb0   = (float*)(ws + OFF_HB0);
  float*    hb1   = (float*)(ws + OFF_HB1);
  v2f*      wrecP = (v2f*)(ws + OFF_WRECP);
  v2f*      winP  = (v2f*)(ws + OFF_WINP);
  unsigned* bar   = (unsigned*)(ws + OFF_BAR);

  // Phase 0: pack weights, copy h_init, reset barrier (524288 threads).
  liquid_prep<<<2048, 256, 0, stream>>>(h0, w_rec, w_in, hb0, wrecP, winP, bar);

  // Phase 1: input projection, 2048 M-tiles x 8 col-groups.
  liquid_input_proj<<<16384, 256, 0, stream>>>(x, winP, bias, xw);

  // Phase 2: persistent sequential scan (32 blocks resident, grid barrier).
  liquid_scan<<<32, 256, 0, stream>>>(xw, wrecP, tau, hb0, hb1,
                                      (float*)d_out, bar);
}